// GNNModel_21595095564867
// MI455X (gfx1250) — compile-verified
//
#include <hip/hip_runtime.h>
#include <hip/hip_bf16.h>

typedef __attribute__((ext_vector_type(2))) float v2f;
typedef __attribute__((ext_vector_type(8))) float v8f;

// ---------------------------------------------------------------------------
// Utility: zero-fill
// ---------------------------------------------------------------------------
__global__ void fill_zero_kernel(float* __restrict__ p, long long n) {
    long long i = (long long)blockIdx.x * blockDim.x + threadIdx.x;
    long long stride = (long long)gridDim.x * blockDim.x;
    for (; i < n; i += stride) p[i] = 0.0f;
}

// ---------------------------------------------------------------------------
// Degree accumulation: outdeg[src[e]] += 1, indeg[dst[e]] += 1
// ---------------------------------------------------------------------------
__global__ void degree_kernel(const int* __restrict__ src, const int* __restrict__ dst,
                              float* __restrict__ outdeg, float* __restrict__ indeg, int E) {
    int e = blockIdx.x * blockDim.x + threadIdx.x;
    if (e < E) {
        atomicAdd(&outdeg[src[e]], 1.0f);
        atomicAdd(&indeg[dst[e]], 1.0f);
    }
}

// In-place deg -> rsqrt(max(deg, 1))
__global__ void invsqrt_kernel(float* __restrict__ deg, int n) {
    int i = blockIdx.x * blockDim.x + threadIdx.x;
    if (i < n) deg[i] = rsqrtf(fmaxf(deg[i], 1.0f));
}

// ---------------------------------------------------------------------------
// C[m0:m0+32, n0:n0+32] = (A[m0:m0+32, :] @ B[:, n0:n0+32]) * rowscale[rows]
// One wave computes a 32x32 fp32 tile as 2x2 WMMA subtiles (128 x
// V_WMMA_F32_16X16X4_F32 per wave).  K and NC are compile-time so every load
// is a single global_load with an immediate offset from 4 base pointers.
// Requires M % 32 == 0 (100000 = 3125*32), NC % 32 == 0, K % 4 == 0.
// VGPR layouts per CDNA5 ISA 7.12.2:
//   A (16x4):  lane&15 = M row, lane>>4 selects K pair {0,1} vs {2,3}
//   B (4x16):  lane&15 = N col, lane>>4 selects K pair
//   C/D:       VGPR j holds row (j + 8*(lane>>4)), col = lane&15
// ---------------------------------------------------------------------------
template <int K, int NC>
__global__ __launch_bounds__(32) void gemm_rowscale_wmma(
    const float* __restrict__ A, const float* __restrict__ B,
    const float* __restrict__ rowscale, float* __restrict__ C) {
    const int lane = threadIdx.x;
    const int half = lane >> 4;   // 0 or 1 -> K pair select
    const int l    = lane & 15;
    const int m0   = blockIdx.x * 32;
    const int n0   = blockIdx.y * 32;

    // Base pointers; all subsequent accesses are compile-time immediates.
    const float* Arow0 = A + (size_t)(m0 + l) * K + half * 2;  // rows m0+l
    const float* Arow1 = Arow0 + (size_t)16 * K;               // rows m0+16+l
    const float* Bc0   = B + (size_t)(half * 2) * NC + (n0 + l);       // cols n0+l
    const float* Bc1   = Bc0 + 16;                                     // cols n0+16+l

    v8f acc[2][2] = {};
#pragma unroll
    for (int k0 = 0; k0 < K; k0 += 4) {
        v2f a0, a1, b0, b1;
        a0.x = Arow0[k0];
        a0.y = Arow0[k0 + 1];
        a1.x = Arow1[k0];
        a1.y = Arow1[k0 + 1];
        b0.x = Bc0[k0 * NC];
        b0.y = Bc0[k0 * NC + NC];
        b1.x = Bc1[k0 * NC];
        b1.y = Bc1[k0 * NC + NC];
        // 8 args: (neg_a, A, neg_b, B, c_mod, C, reuse_a, reuse_b)
        acc[0][0] = __builtin_amdgcn_wmma_f32_16x16x4_f32(
            false, a0, false, b0, (short)0, acc[0][0], false, false);
        acc[0][1] = __builtin_amdgcn_wmma_f32_16x16x4_f32(
            false, a0, false, b1, (short)0, acc[0][1], false, false);
        acc[1][0] = __builtin_amdgcn_wmma_f32_16x16x4_f32(
            false, a1, false, b0, (short)0, acc[1][0], false, false);
        acc[1][1] = __builtin_amdgcn_wmma_f32_16x16x4_f32(
            false, a1, false, b1, (short)0, acc[1][1], false, false);
    }

#pragma unroll
    for (int mt = 0; mt < 2; ++mt) {
#pragma unroll
        for (int j = 0; j < 8; ++j) {
            const int m = m0 + mt * 16 + j + 8 * half;
            const float rs = rowscale[m];
            float* crow = C + (size_t)m * NC + n0 + l;
            crow[0]  = acc[mt][0][j] * rs;
            crow[16] = acc[mt][1][j] * rs;
        }
    }
}

// ---------------------------------------------------------------------------
// SpMM scatter: AGG[dst[e], :] += H[src[e], :]
// One wave per edge; each lane owns VEC contiguous features (F == 32*VEC).
// ---------------------------------------------------------------------------
template <int VEC>
__global__ void spmm_atomic_kernel(const int* __restrict__ src, const int* __restrict__ dst,
                                   const float* __restrict__ H, float* __restrict__ AGG,
                                   int E, int F) {
    long long tid = (long long)blockIdx.x * blockDim.x + threadIdx.x;
    int e    = (int)(tid >> 5);
    int lane = (int)(tid & 31);
    if (e >= E) return;
    const int s = src[e];
    const int d = dst[e];
    const float* hs = H + (size_t)s * F + lane * VEC;
    float* ag       = AGG + (size_t)d * F + lane * VEC;
    float vals[VEC];
#pragma unroll
    for (int v = 0; v < VEC; ++v) vals[v] = hs[v];
#pragma unroll
    for (int v = 0; v < VEC; ++v) atomicAdd(&ag[v], vals[v]);
}

// ---------------------------------------------------------------------------
// out[i] = (in[i] * invdeg[row] + bias[col]) , optional ReLU. May be in-place.
// ---------------------------------------------------------------------------
__global__ void scale_bias_act_kernel(const float* in, float* out,
                                      const float* __restrict__ invdeg,
                                      const float* __restrict__ bias,
                                      int total, int F, int do_relu) {
    int i = blockIdx.x * blockDim.x + threadIdx.x;
    if (i >= total) return;
    int row = i / F;
    int col = i - row * F;
    float v = in[i] * invdeg[row] + bias[col];
    out[i] = do_relu ? fmaxf(v, 0.0f) : v;
}

// ---------------------------------------------------------------------------
// Host-side orchestration
// ---------------------------------------------------------------------------
extern "C" void kernel_launch(void* const* d_in, const int* in_sizes, int n_in,
                              void* d_out, int out_size, void* d_ws, size_t ws_size,
                              hipStream_t stream) {
    const float* features = (const float*)d_in[0];
    const int*   src      = (const int*)d_in[1];
    const int*   dst      = (const int*)d_in[2];
    const float* W1       = (const float*)d_in[3];
    const float* b1       = (const float*)d_in[4];
    const float* W2       = (const float*)d_in[5];
    const float* b2       = (const float*)d_in[6];
    float*       outp     = (float*)d_out;

    const int D_IN = 128, D_HID = 128, D_OUT = 64;
    const int N = in_sizes[0] / D_IN;   // 100000
    const int E = in_sizes[1];          // 3200000

    // Workspace layout (floats):
    //   [0, N)             outdeg -> inv_sqrt_out
    //   [N, 2N)            indeg  -> inv_sqrt_in
    //   [2N, 2N+128N)      H1   (layer-1 GEMM output / SpMM source)
    //   [2N+128N, 2N+256N) AGG1 (layer-1 aggregation; becomes layer-2 input)
    // After layer 1, the H1 region is recycled:
    //   G    = H1 base        (N x 64, layer-2 GEMM output)
    //   AGG2 = H1 base + 64N  (N x 64, layer-2 aggregation)
    float* ws     = (float*)d_ws;
    float* outdeg = ws;
    float* indeg  = ws + N;
    float* H1     = ws + 2 * (size_t)N;
    float* AGG1   = H1 + (size_t)N * D_HID;
    float* G      = H1;
    float* AGG2   = H1 + (size_t)N * D_OUT;

    // 1) degrees
    fill_zero_kernel<<<256, 256, 0, stream>>>(outdeg, 2LL * N);
    degree_kernel<<<(E + 255) / 256, 256, 0, stream>>>(src, dst, outdeg, indeg, E);
    invsqrt_kernel<<<(2 * N + 255) / 256, 256, 0, stream>>>(outdeg, 2 * N);

    // 2) layer 1: H1 = (X @ W1) * inv_sqrt_out
    dim3 g1(N / 32, D_HID / 32);
    gemm_rowscale_wmma<128, 128><<<g1, 32, 0, stream>>>(features, W1, outdeg, H1);

    // 3) SpMM 1: AGG1[dst] += H1[src]
    fill_zero_kernel<<<2048, 256, 0, stream>>>(AGG1, (long long)N * D_HID);
    long long th1 = (long long)E * 32;
    spmm_atomic_kernel<4><<<(unsigned)((th1 + 255) / 256), 256, 0, stream>>>(
        src, dst, H1, AGG1, E, D_HID);

    // 4) AGG1 = relu(AGG1 * inv_sqrt_in + b1)   (in place -> layer-2 input)
    int tot1 = N * D_HID;
    scale_bias_act_kernel<<<(tot1 + 255) / 256, 256, 0, stream>>>(
        AGG1, AGG1, indeg, b1, tot1, D_HID, 1);

    // 5) layer 2: G = (AGG1 @ W2) * inv_sqrt_out
    dim3 g2(N / 32, D_OUT / 32);
    gemm_rowscale_wmma<128, 64><<<g2, 32, 0, stream>>>(AGG1, W2, outdeg, G);

    // 6) SpMM 2: AGG2[dst] += G[src]
    fill_zero_kernel<<<2048, 256, 0, stream>>>(AGG2, (long long)N * D_OUT);
    spmm_atomic_kernel<2><<<(unsigned)((th1 + 255) / 256), 256, 0, stream>>>(
        src, dst, G, AGG2, E, D_OUT);

    // 7) out = AGG2 * inv_sqrt_in + b2
    int tot2 = N * D_OUT;
    scale_bias_act_kernel<<<(tot2 + 255) / 256, 256, 0, stream>>>(
        AGG2, outp, indeg, b2, tot2, D_OUT, 0);
}